// MR_Forward_Model_Static_24111946400509
// MI455X (gfx1250) — compile-verified
//
#include <hip/hip_runtime.h>

// MR forward model: coil encode + separable type-2 NUDFT as f32 WMMA GEMM.
// T=2, CH=8, D=4, H=W=128, K=2048.
// Karatsuba complex product: 3 WMMAs per 16x16x4 chunk instead of 4.
// KT=32: two k-sets per A-fragment load to amortize global traffic.
typedef __attribute__((ext_vector_type(2))) float v2f;
typedef __attribute__((ext_vector_type(8))) float v8f;

#define T_  2
#define CH_ 8
#define D_  4
#define H_  128
#define W_  128
#define K_  2048
#define KT  32            // k-space columns per workgroup (2 WMMA N-tiles)
#define EX_STRIDE 132     // [k][x] rows padded: banks (4k + x) -> conflict-free
#define EY_STRIDE 33      // [y][k] rows padded

#define WMMA_F32(A, B, C) \
    __builtin_amdgcn_wmma_f32_16x16x4_f32(false, (A), false, (B), (short)0, (C), false, false)

__global__ __launch_bounds__(128)
void mr_fwd_nudft_wmma(const float* __restrict__ img_r,
                       const float* __restrict__ img_i,
                       const float* __restrict__ csm_r,
                       const float* __restrict__ csm_i,
                       const float* __restrict__ traj,
                       float* __restrict__ out)
{
    __shared__ float ExC[KT * EX_STRIDE];   // cos(kx*x)
    __shared__ float ExS[KT * EX_STRIDE];   // sin(kx*x)
    __shared__ float ExD[KT * EX_STRIDE];   // cos - sin (Karatsuba third term)
    __shared__ float EyC[H_ * EY_STRIDE];   // cos(ky*y)
    __shared__ float EyS[H_ * EY_STRIDE];   // sin(ky*y)

    const int ktile = blockIdx.x;           // 0..K_/KT-1
    const int ch    = blockIdx.y;           // 0..CH_-1
    const int t     = blockIdx.z;           // 0..T_-1
    const int tid   = threadIdx.x;          // 0..127
    const int kbase = ktile * KT;

    // ---- twiddle fill: 2*KT*128 sincos per workgroup (separable phase) ----
    const float* trj_ky = traj + ((size_t)t * 2 + 0) * K_ + kbase; // multiplies y
    const float* trj_kx = traj + ((size_t)t * 2 + 1) * K_ + kbase; // multiplies x
    for (int idx = tid; idx < KT * W_; idx += 128) {
        const int k = idx >> 7;            // /128
        const int x = idx & 127;
        const float p = trj_kx[k] * (float)(x - W_ / 2);
        float s, c; __sincosf(p, &s, &c);
        ExC[k * EX_STRIDE + x] = c;
        ExS[k * EX_STRIDE + x] = s;
        ExD[k * EX_STRIDE + x] = c - s;
    }
    for (int idx = tid; idx < KT * H_; idx += 128) {
        const int k = idx >> 7;
        const int y = idx & 127;
        const float p = trj_ky[k] * (float)(y - H_ / 2);
        float s, c; __sincosf(p, &s, &c);
        EyC[y * EY_STRIDE + k] = c;
        EyS[y * EY_STRIDE + k] = s;
    }
    __syncthreads();

    // ---- per-wave complex GEMM over x, fused Ey reduction over y ----
    const int lane = tid & 31;
    const int d    = tid >> 5;             // wave id -> depth slice (D_=4 waves)
    const int m    = lane & 15;            // A row (y) / B,D column within k-set
    const int hi   = lane >> 4;            // K-dim half select per ISA layout

    const size_t imgBase = ((size_t)(t * D_ + d)) * (size_t)(H_ * W_);
    const size_t csmBase = ((size_t)((t * CH_ + ch) * D_ + d)) * (size_t)(H_ * W_);

    float accR0 = 0.f, accI0 = 0.f;        // k-set 0: columns kbase + [0,16)
    float accR1 = 0.f, accI1 = 0.f;        // k-set 1: columns kbase + [16,32)

    for (int yt = 0; yt < H_; yt += 16) {
        // Karatsuba partial products (per k-set):
        //   M1 += Ar*c ; M2 += Ai*s ; M3 += (Ar+Ai)*(c-s)
        //   Dr = M1+M2 ; Di = M3-M1+M2
        v8f M1a = {}, M2a = {}, M3a = {};
        v8f M1b = {}, M2b = {}, M3b = {};

        const int y = yt + m;              // A-matrix row, both lane halves
        const float* ir = img_r + imgBase + (size_t)y * W_;
        const float* ii = img_i + imgBase + (size_t)y * W_;
        const float* cr = csm_r + csmBase + (size_t)y * W_;
        const float* ci = csm_i + csmBase + (size_t)y * W_;

        #pragma unroll 4
        for (int xc = 0; xc < W_; xc += 4) {
            const int x0 = xc + 2 * hi;    // lanes 0-15: K0,K1; lanes 16-31: K2,K3
            // A fragment: multi = image * csm (complex), two x-consecutive elems
            const v2f a_ir = *(const v2f*)(ir + x0);
            const v2f a_ii = *(const v2f*)(ii + x0);
            const v2f a_cr = *(const v2f*)(cr + x0);
            const v2f a_ci = *(const v2f*)(ci + x0);
            const v2f Ar = a_ir * a_cr - a_ii * a_ci;
            const v2f Ai = a_ir * a_ci + a_ii * a_cr;
            const v2f As = Ar + Ai;

            // B fragments, k-set 0: rows (reduction x) = x0, x0+1, column = m
            const v2f Bc0 = *(const v2f*)(&ExC[m * EX_STRIDE + x0]);
            const v2f Bs0 = *(const v2f*)(&ExS[m * EX_STRIDE + x0]);
            const v2f Bd0 = *(const v2f*)(&ExD[m * EX_STRIDE + x0]);
            M1a = WMMA_F32(Ar, Bc0, M1a);
            M2a = WMMA_F32(Ai, Bs0, M2a);
            M3a = WMMA_F32(As, Bd0, M3a);

            // B fragments, k-set 1: column = m + 16
            const v2f Bc1 = *(const v2f*)(&ExC[(m + 16) * EX_STRIDE + x0]);
            const v2f Bs1 = *(const v2f*)(&ExS[(m + 16) * EX_STRIDE + x0]);
            const v2f Bd1 = *(const v2f*)(&ExD[(m + 16) * EX_STRIDE + x0]);
            M1b = WMMA_F32(Ar, Bc1, M1b);
            M2b = WMMA_F32(Ai, Bs1, M2b);
            M3b = WMMA_F32(As, Bd1, M3b);
        }

        // Epilogue: reconstruct complex tile, apply Ey = cos - i*sin, reduce
        // over tile rows. C/D layout: VGPR j, lane l -> row M=j+8*hi, col N=l&15.
        #pragma unroll
        for (int j = 0; j < 8; ++j) {
            const int yy = yt + j + 8 * hi;
            {   // k-set 0
                const float cy = EyC[yy * EY_STRIDE + m];
                const float sy = EyS[yy * EY_STRIDE + m];
                const float dr = M1a[j] + M2a[j];
                const float di = M3a[j] - M1a[j] + M2a[j];
                accR0 += dr * cy + di * sy;
                accI0 += di * cy - dr * sy;
            }
            {   // k-set 1
                const float cy = EyC[yy * EY_STRIDE + m + 16];
                const float sy = EyS[yy * EY_STRIDE + m + 16];
                const float dr = M1b[j] + M2b[j];
                const float di = M3b[j] - M1b[j] + M2b[j];
                accR1 += dr * cy + di * sy;
                accI1 += di * cy - dr * sy;
            }
        }
    }

    // Fold the two lane halves (disjoint y row sets, same k column).
    accR0 += __shfl_xor(accR0, 16, 32);
    accI0 += __shfl_xor(accI0, 16, 32);
    accR1 += __shfl_xor(accR1, 16, 32);
    accI1 += __shfl_xor(accI1, 16, 32);

    if (hi == 0) {
        const int n = ch * D_ + d;                         // matches reshape(ch*d)
        const size_t base = ((size_t)(t * CH_ * D_ + n)) * K_;
        {
            const size_t o = (base + (kbase + m)) * 2;     // complex64 interleaved
            v2f w; w.x = accR0; w.y = accI0;
            *(v2f*)(out + o) = w;
        }
        {
            const size_t o = (base + (kbase + 16 + m)) * 2;
            v2f w; w.x = accR1; w.y = accI1;
            *(v2f*)(out + o) = w;
        }
    }
}

extern "C" void kernel_launch(void* const* d_in, const int* in_sizes, int n_in,
                              void* d_out, int out_size, void* d_ws, size_t ws_size,
                              hipStream_t stream) {
    (void)in_sizes; (void)n_in; (void)out_size; (void)d_ws; (void)ws_size;
    const float* img_r = (const float*)d_in[0];
    const float* img_i = (const float*)d_in[1];
    const float* csm_r = (const float*)d_in[2];
    const float* csm_i = (const float*)d_in[3];
    const float* traj  = (const float*)d_in[4];
    float* out = (float*)d_out;

    dim3 grid(K_ / KT, CH_, T_);   // 64 x 8 x 2 = 1024 workgroups
    dim3 block(128);               // 4 waves: one per depth slice d
    mr_fwd_nudft_wmma<<<grid, block, 0, stream>>>(img_r, img_i, csm_r, csm_i,
                                                  traj, out);
}